// Instance_block_66924180406903
// MI455X (gfx1250) — compile-verified
//
#include <hip/hip_runtime.h>
#include <hip/hip_bf16.h>

// ---------------------------------------------------------------------------
// Matching-pursuit style update on MI455X (gfx1250, wave32):
//   resid = Y - phi @ X[:, :k]                 (GEMM1, 64x256 * 256xk)
//   upd   = X[:, :k] + step * (phi^T @ resid)  (GEMM2, 256x64 * 64xk)
//   X[:, :k] = where(|upd| > 0.1, upd, 0)
// Output = [phi (64*256), X (256*131072)] flat, fp32.
//
// fp32 WMMA path: V_WMMA_F32_16X16X4_F32 (D = A(16x4) x B(4x16) + C).
// Per-wave: 16 output columns; 8 waves -> 128 columns per block.
// phi staged once in LDS transposed with stride-65 padding; serves A-frags of
// both GEMMs conflict-free-ish. resid staged wave-privately in LDS (stride 20)
// to convert WMMA D-layout -> B-layout between the GEMMs.
// ---------------------------------------------------------------------------

typedef __attribute__((ext_vector_type(2))) float v2f;
typedef __attribute__((ext_vector_type(8))) float v8f;

#define NCOLS   131072u   // N
#define KDIM    256u      // inner dim of GEMM1 / rows of X
#define MROWS   64u       // rows of phi / Y
#define PHI_N   16384u    // 64*256
#define THRESH  0.1f

#define PHIT_STRIDE 65u                    // padded row stride (floats) of phi^T in LDS
#define PHIT_ELEMS  (KDIM * PHIT_STRIDE)   // 16640 floats
#define RES_STRIDE  20u                    // padded row stride of resid tile in LDS
#define RES_ELEMS   (MROWS * RES_STRIDE)   // 1280 floats per wave
#define LDS_FLOATS  (PHIT_ELEMS + 8u * RES_ELEMS)  // 26880 floats = 107520 B

__global__ void __launch_bounds__(256)
omp_update_kernel(const float* __restrict__ phi,
                  const float* __restrict__ X,
                  const float* __restrict__ Y,
                  const float* __restrict__ step,
                  const int*   __restrict__ idx,
                  float*       __restrict__ out)
{
    extern __shared__ float lds[];
    float* ldsPhiT = lds;                                   // phiT[k][m] @ k*65+m
    const unsigned tid    = threadIdx.x;
    const unsigned waveId = tid >> 5;
    const unsigned lane   = tid & 31u;
    const unsigned lane16 = lane & 15u;
    const unsigned half   = lane >> 4;                      // 0 or 1
    float* ldsRes = lds + PHIT_ELEMS + waveId * RES_ELEMS;  // wave-private

    const unsigned k  = (unsigned)(idx[0] + 1);             // active columns
    const unsigned blockCol0 = blockIdx.x * 128u;
    float* outX = out + PHI_N;                              // X region of output

    // ---- pure passthrough block (all 128 cols >= k): copy X -> out ----
    if (blockCol0 >= k) {
        const unsigned col = blockCol0 + (tid & 127u);
        for (unsigned r = tid >> 7; r < KDIM; r += 2u) {
            const unsigned off = r * NCOLS + col;
            outX[off] = X[off];
        }
        return;
    }

    // ---- cooperative load of phi into LDS, transposed + padded ----
    for (unsigned e = tid; e < PHI_N; e += 256u) {
        const unsigned m  = e >> 8;     // 0..63
        const unsigned kk = e & 255u;   // 0..255
        ldsPhiT[kk * PHIT_STRIDE + m] = phi[e];   // lane-distinct banks (stride 65)
    }
    __syncthreads();

    const float st  = step[0];
    const unsigned col = blockCol0 + waveId * 16u + lane16; // this lane's column (B/C/D role)

    // ================= GEMM1: acc = phi @ Xcols  (64 x 16 per wave) =========
    // A 16x4 layout (ISA): lanes0-15 hold M=lane, V0=K0/V1=K1; lanes16-31 K2/K3.
    // B 4x16 assumed mirror: V0 = {K0 | K2}, V1 = {K1 | K3}, N = lane%16.
    v8f acc[4];
    #pragma unroll
    for (int t = 0; t < 4; ++t) acc[t] = (v8f)0.0f;

    for (unsigned kk = 0; kk < KDIM; kk += 4u) {
        const unsigned rb = kk + 2u * half;
        v2f b;
        b.x = X[rb * NCOLS + col];
        b.y = X[(rb + 1u) * NCOLS + col];
        #pragma unroll
        for (unsigned t = 0; t < 4; ++t) {
            const unsigned m = t * 16u + lane16;
            v2f a;
            a.x = ldsPhiT[(rb)      * PHIT_STRIDE + m];   // phi[m][rb]
            a.y = ldsPhiT[(rb + 1u) * PHIT_STRIDE + m];   // phi[m][rb+1]
            acc[t] = __builtin_amdgcn_wmma_f32_16x16x4_f32(
                /*neg_a=*/false, a, /*neg_b=*/false, b,
                /*c_mod=*/(short)0, acc[t],
                /*reuse_a=*/false, /*reuse_b=*/false);
        }
    }

    // ---- resid = Y - acc ; stage to wave-private LDS (D-layout -> rows) ----
    #pragma unroll
    for (unsigned t = 0; t < 4; ++t) {
        #pragma unroll
        for (unsigned i = 0; i < 8; ++i) {
            const unsigned r = t * 16u + i + 8u * half;   // D-layout row of VGPR i
            const float yv = Y[r * NCOLS + col];
            ldsRes[r * RES_STRIDE + lane16] = yv - acc[t][i];
        }
    }
    // wave-private LDS: in-wave DS ordering only (compiler inserts s_wait_dscnt)

    // ================= GEMM2: delta = phi^T @ resid (256 x 16 per wave) =====
    for (unsigned mt = 0; mt < 16u; ++mt) {
        v8f acc2 = (v8f)0.0f;
        for (unsigned kk = 0; kk < MROWS; kk += 4u) {
            const unsigned c0 = kk + 2u * half;
            v2f a, b;
            // A' = phi^T : A'[m'][k'] = phi[k'][m'] = ldsPhiT[m'*65 + k']
            const unsigned mp = mt * 16u + lane16;
            a.x = ldsPhiT[mp * PHIT_STRIDE + c0];
            a.y = ldsPhiT[mp * PHIT_STRIDE + c0 + 1u];
            // B = resid rows kk..kk+3
            b.x = ldsRes[(c0)      * RES_STRIDE + lane16];
            b.y = ldsRes[(c0 + 1u) * RES_STRIDE + lane16];
            acc2 = __builtin_amdgcn_wmma_f32_16x16x4_f32(
                false, a, false, b, (short)0, acc2, false, false);
        }
        // ---- epilogue: upd = X + step*delta ; threshold ; masked store ----
        #pragma unroll
        for (unsigned i = 0; i < 8; ++i) {
            const unsigned r   = mt * 16u + i + 8u * half;
            const unsigned off = r * NCOLS + col;
            const float xv = X[off];
            float u = fmaf(st, acc2[i], xv);
            u = (__builtin_fabsf(u) > THRESH) ? u : 0.0f;
            outX[off] = (col < k) ? u : xv;   // boundary block: passthrough cols >= k
        }
    }
}

// ---- copy phi (64*256 floats) to the head of the output buffer ----
__global__ void __launch_bounds__(256)
phi_copy_kernel(const float* __restrict__ phi, float* __restrict__ out)
{
    const unsigned i = blockIdx.x * 256u + threadIdx.x;
    if (i < PHI_N) out[i] = phi[i];
}

extern "C" void kernel_launch(void* const* d_in, const int* in_sizes, int n_in,
                              void* d_out, int out_size, void* d_ws, size_t ws_size,
                              hipStream_t stream)
{
    const float* phi  = (const float*)d_in[0];
    const float* X    = (const float*)d_in[1];
    const float* Y    = (const float*)d_in[2];
    const float* step = (const float*)d_in[3];
    const int*   idx  = (const int*)d_in[4];
    float* out = (float*)d_out;

    const size_t ldsBytes = LDS_FLOATS * sizeof(float);   // 107520 B (>64KB -> raise cap)
    (void)hipFuncSetAttribute((const void*)omp_update_kernel,
                              hipFuncAttributeMaxDynamicSharedMemorySize,
                              (int)ldsBytes);

    phi_copy_kernel<<<(PHI_N + 255u) / 256u, 256, 0, stream>>>(phi, out);

    const unsigned nBlocks = NCOLS / 128u;                // 1024 blocks, 8 waves each
    omp_update_kernel<<<nBlocks, 256, ldsBytes, stream>>>(phi, X, Y, step, idx, out);
}